// Triangle_Att_Start_34660386079028
// MI455X (gfx1250) — compile-verified
//
#include <hip/hip_runtime.h>

typedef float v2f __attribute__((ext_vector_type(2)));
typedef float v4f __attribute__((ext_vector_type(4)));
typedef float v8f __attribute__((ext_vector_type(8)));

#define LDIM 256
#define PDIM 128
#define NPOS 65536          // L*L
#define NHCOL 64            // num_heads * hidden
#define NHEADS 4
#define HD 16
#define EPSV 1e-5f

__device__ __forceinline__ float sig_(float x) { return 1.0f / (1.0f + __expf(-x)); }

// ---------------- kernel 0: zero groupnorm stats ----------------
__global__ void k0_zero(float* __restrict__ g) {
    if (threadIdx.x < 8) g[threadIdx.x] = 0.0f;
}

// ---------------- kernel 1: LayerNorm + Q/K/V/G/Wb projections (WMMA fp32) ----------------
__global__ __launch_bounds__(256) void k1_ln_proj(
    const float* __restrict__ pair, const float* __restrict__ ln_w, const float* __restrict__ ln_b,
    const float* __restrict__ Wq, const float* __restrict__ Wk, const float* __restrict__ Wv,
    const float* __restrict__ Wb, const float* __restrict__ Wg,
    float* __restrict__ xout, float* __restrict__ Qb, float* __restrict__ Kb,
    float* __restrict__ Vb, float* __restrict__ Gb, float* __restrict__ Bb)
{
    __shared__ float xs[16 * 132];                 // 16 rows x 128 cols, stride 132 (bank-conflict free)
    const int t    = threadIdx.x;
    const int tile = blockIdx.x;                   // 4096 tiles of 16 positions
    const int r    = t >> 4;                       // row 0..15
    const int sub  = t & 15;                       // 16 threads per row, 8 channels each
    const int pos  = tile * 16 + r;

    // ---- LayerNorm over 128 channels ----
    float v[8];
    float s1 = 0.0f, s2 = 0.0f;
    const float* prow = pair + (size_t)pos * PDIM + sub * 8;
    #pragma unroll
    for (int u = 0; u < 8; ++u) {
        v[u] = __builtin_nontemporal_load(&prow[u]);   // read-once input: TH=NT
        s1 += v[u]; s2 += v[u] * v[u];
    }
    #pragma unroll
    for (int m = 1; m < 16; m <<= 1) {             // reduce across 16 lanes (stays in half-wave)
        s1 += __shfl_xor(s1, m, 32);
        s2 += __shfl_xor(s2, m, 32);
    }
    const float mean = s1 * (1.0f / 128.0f);
    const float var  = s2 * (1.0f / 128.0f) - mean * mean;
    const float rstd = rsqrtf(var + EPSV);
    #pragma unroll
    for (int u = 0; u < 8; ++u) {
        const int c  = sub * 8 + u;
        const float x = (v[u] - mean) * rstd * ln_w[c] + ln_b[c];
        xs[r * 132 + c] = x;
        __builtin_nontemporal_store(x, &xout[(size_t)pos * PDIM + c]);  // write-once output
    }
    __syncthreads();

    // ---- Wb projection: 4 columns, scalar dots ----
    if (t < 64) {
        const int rr = t >> 2, cb = t & 3;
        float acc = 0.0f;
        #pragma unroll 4
        for (int p = 0; p < PDIM; ++p) acc += xs[rr * 132 + p] * Wb[p * 4 + cb];
        Bb[(size_t)(tile * 16 + rr) * 4 + cb] = acc;
    }

    // ---- WMMA: x[16x128] @ W[128x16] per N-tile; 16 tiles (Q,K,V,G x 4) over 8 waves ----
    const int w    = t >> 5;
    const int lane = t & 31;
    const int half = lane >> 4;
    const int l16  = lane & 15;
    #pragma unroll
    for (int tt = 0; tt < 2; ++tt) {
        const int tn = w + tt * 8;                 // 0..15
        const float* Ws; float* dst;
        switch (tn >> 2) {
            case 0:  Ws = Wq; dst = Qb; break;
            case 1:  Ws = Wk; dst = Kb; break;
            case 2:  Ws = Wv; dst = Vb; break;
            default: Ws = Wg; dst = Gb; break;
        }
        const int nc0 = (tn & 3) * 16;
        v8f c = {};
        for (int k0 = 0; k0 < PDIM; k0 += 4) {
            v2f a, b;
            // A fragment (16x4): lanes 0-15 hold K={k0,k0+1}, lanes 16-31 K={k0+2,k0+3}
            a.x = xs[l16 * 132 + k0 + half * 2 + 0];
            a.y = xs[l16 * 132 + k0 + half * 2 + 1];
            // B fragment (4x16): rows striped across lanes (weights stay L2-resident)
            b.x = Ws[(size_t)(k0 + half * 2 + 0) * NHCOL + nc0 + l16];
            b.y = Ws[(size_t)(k0 + half * 2 + 1) * NHCOL + nc0 + l16];
            c = __builtin_amdgcn_wmma_f32_16x16x4_f32(false, a, false, b, (short)0, c, false, false);
        }
        #pragma unroll
        for (int vg = 0; vg < 8; ++vg) {           // C/D: VGPR r -> M=r (lanes<16) / M=r+8
            const int m = vg + half * 8;
            dst[(size_t)(tile * 16 + m) * NHCOL + nc0 + l16] = c[vg];  // RT store: re-read by k2/k4 via L2
        }
    }
}

// ---------------- kernel 2: KV outer product + pre-groupnorm out + stats ----------------
__global__ __launch_bounds__(256) void k2_kv(
    const float* __restrict__ Qb, const float* __restrict__ Kb, const float* __restrict__ Vb,
    const float* __restrict__ Bb,
    float* __restrict__ kvout, float* __restrict__ outpre, float* __restrict__ gstats)
{
    __shared__ float vsh[16 * 64];
    __shared__ float bT[64];
    __shared__ float sred[8];
    const int t    = threadIdx.x;
    const int tile = blockIdx.x;                   // 16 positions per block
    const int pl   = t >> 4;
    const int k    = t & 15;
    const int pos  = tile * 16 + pl;

    #pragma unroll
    for (int it = 0; it < 4; ++it) {
        const int idx = t + it * 256;
        vsh[idx] = Vb[(size_t)tile * 1024 + idx];
    }
    if (t < 64) {                                  // transposed bias projection b[j,i,n]
        const int pp = t >> 2, n = t & 3;
        const int pg = tile * 16 + pp;
        const int i = pg >> 8, j = pg & 255;
        bT[t] = Bb[(size_t)(j * 256 + i) * 4 + n];
    }
    if (t < 8) sred[t] = 0.0f;
    __syncthreads();

    float lsum[NHEADS], lsq[NHEADS];
    #pragma unroll
    for (int n = 0; n < NHEADS; ++n) {
        const float Kv = Kb[(size_t)pos * NHCOL + n * HD + k];
        const float Qv = Qb[(size_t)pos * NHCOL + n * HD + k];
        float s = Kv * Qv;
        #pragma unroll
        for (int m = 1; m < 16; m <<= 1) s += __shfl_xor(s, m, 32);

        // current_kv[n,pos,k,0:16] = K[k] * V[:]  -- 268MB write-once stream: TH=NT,
        // keeps the 192MB L2 free for Q/K/V/G/outpre producer->consumer traffic.
        const size_t base = ((size_t)n * NPOS + pos) * 256 + (size_t)k * 16;
        const float* vr = &vsh[pl * 64 + n * HD];
        v4f* kv4 = (v4f*)(kvout + base);
        #pragma unroll
        for (int q = 0; q < 4; ++q) {
            v4f rr;
            rr.x = Kv * vr[q * 4 + 0]; rr.y = Kv * vr[q * 4 + 1];
            rr.z = Kv * vr[q * 4 + 2]; rr.w = Kv * vr[q * 4 + 3];
            __builtin_nontemporal_store(rr, &kv4[q]);
        }
        // out_pre[n,pos,v=k] = (Q.K + bproj[j,i,n]) * V[v]
        const float op = (s + bT[pl * 4 + n]) * vr[k];
        outpre[((size_t)n * NPOS + pos) * HD + k] = op;   // RT: re-read by k4
        lsum[n] = op; lsq[n] = op * op;
    }
    #pragma unroll
    for (int n = 0; n < NHEADS; ++n) {
        atomicAdd(&sred[n], lsum[n]);
        atomicAdd(&sred[4 + n], lsq[n]);
    }
    __syncthreads();
    if (t < 8) atomicAdd(&gstats[t], sred[t]);
}

// ---------------- kernel 4: groupnorm + gating + output projection (WMMA fp32) ----------------
__global__ __launch_bounds__(256) void k4_out(
    const float* __restrict__ outpre, const float* __restrict__ Gb, const float* __restrict__ bg,
    const float* __restrict__ Wo, const float* __restrict__ bo, const float* __restrict__ gstats,
    float* __restrict__ yout)
{
    __shared__ float ysh[16 * 68];                 // stride 68 -> conflict-free column reads
    const int t    = threadIdx.x;
    const int tile = blockIdx.x;
    const float cntinv = 1.0f / 1048576.0f;        // 256*256*16 elements per head
    float mean[NHEADS], rstd[NHEADS];
    #pragma unroll
    for (int n = 0; n < NHEADS; ++n) {
        const float m  = gstats[n] * cntinv;
        const float va = gstats[4 + n] * cntinv - m * m;
        mean[n] = m; rstd[n] = rsqrtf(va + EPSV);
    }
    #pragma unroll
    for (int it = 0; it < 4; ++it) {
        const int idx = t + it * 256;
        const int pl = idx >> 6, c = idx & 63;
        const int n = c >> 4, h = c & 15;
        const int pos = tile * 16 + pl;
        const float op = outpre[((size_t)n * NPOS + pos) * HD + h];
        float y = (op - mean[n]) * rstd[n];
        const float g = sig_(Gb[(size_t)pos * NHCOL + c] + bg[c]);
        y *= g * sig_(g);
        ysh[pl * 68 + c] = y;
    }
    __syncthreads();

    // y[16x64] @ Wo[64x128] + bo ; 8 N-tiles, one per wave
    const int w = t >> 5, lane = t & 31, half = lane >> 4, l16 = lane & 15;
    const int nc0 = w * 16;
    v8f c = {};
    for (int k0 = 0; k0 < NHCOL; k0 += 4) {
        v2f a, b;
        a.x = ysh[l16 * 68 + k0 + half * 2 + 0];
        a.y = ysh[l16 * 68 + k0 + half * 2 + 1];
        b.x = Wo[(size_t)(k0 + half * 2 + 0) * PDIM + nc0 + l16];
        b.y = Wo[(size_t)(k0 + half * 2 + 1) * PDIM + nc0 + l16];
        c = __builtin_amdgcn_wmma_f32_16x16x4_f32(false, a, false, b, (short)0, c, false, false);
    }
    #pragma unroll
    for (int vg = 0; vg < 8; ++vg) {
        const int m = vg + half * 8;
        const float val = c[vg] + bo[nc0 + l16];
        __builtin_nontemporal_store(val, &yout[(size_t)(tile * 16 + m) * PDIM + nc0 + l16]);
    }
}

// ---------------- launch ----------------
extern "C" void kernel_launch(void* const* d_in, const int* in_sizes, int n_in,
                              void* d_out, int out_size, void* d_ws, size_t ws_size,
                              hipStream_t stream)
{
    const float* pair = (const float*)d_in[0];
    const float* ln_w = (const float*)d_in[1];
    const float* ln_b = (const float*)d_in[2];
    const float* Wq   = (const float*)d_in[3];
    const float* Wk   = (const float*)d_in[4];
    const float* Wv   = (const float*)d_in[5];
    const float* Wb   = (const float*)d_in[6];
    const float* Wg   = (const float*)d_in[7];
    const float* bg   = (const float*)d_in[8];
    const float* Wo   = (const float*)d_in[9];
    const float* bo   = (const float*)d_in[10];

    // outputs concatenated flat in return order: x, current_kv, out
    float* out   = (float*)d_out;
    float* xout  = out;
    float* kvout = xout + (size_t)NPOS * PDIM;
    float* yout  = kvout + (size_t)NHEADS * NPOS * 256;

    // workspace layout (floats)
    float* ws     = (float*)d_ws;
    float* Qb     = ws;
    float* Kb     = Qb + (size_t)NPOS * NHCOL;
    float* Vb     = Kb + (size_t)NPOS * NHCOL;
    float* Gb     = Vb + (size_t)NPOS * NHCOL;
    float* Bb     = Gb + (size_t)NPOS * NHCOL;
    float* outpre = Bb + (size_t)NPOS * 4;
    float* gstats = outpre + (size_t)NHEADS * NPOS * HD;

    k0_zero<<<1, 32, 0, stream>>>(gstats);
    k1_ln_proj<<<4096, 256, 0, stream>>>(pair, ln_w, ln_b, Wq, Wk, Wv, Wb, Wg,
                                         xout, Qb, Kb, Vb, Gb, Bb);
    k2_kv<<<4096, 256, 0, stream>>>(Qb, Kb, Vb, Bb, kvout, outpre, gstats);
    k4_out<<<4096, 256, 0, stream>>>(outpre, Gb, bg, Wo, bo, gstats, yout);
}